// Encoder_21930103014194
// MI455X (gfx1250) — compile-verified
//
#include <hip/hip_runtime.h>
#include <stdint.h>

typedef float v2f __attribute__((ext_vector_type(2)));
typedef float v8f __attribute__((ext_vector_type(8)));

#define NNODES 50000
#define NEDGES 800000

// ---------------- utility kernels ----------------

__global__ void fill_kernel(float* __restrict__ p, float v, int n) {
    int i = blockIdx.x * blockDim.x + threadIdx.x;
    if (i < n) p[i] = v;
}

__global__ void deg_kernel(const long long* __restrict__ dst, float* __restrict__ deg, int E) {
    int e = blockIdx.x * blockDim.x + threadIdx.x;
    if (e < E) atomicAdd(&deg[(int)dst[e]], 1.0f);
}

__global__ void rsqrt_kernel(float* __restrict__ d, int n) {
    int i = blockIdx.x * blockDim.x + threadIdx.x;
    if (i < n) d[i] = rsqrtf(d[i]);
}

// ---------------- WMMA f32 GEMM: out[M,Nc] = X[M,128] @ W[128,Nc] ----------------
// One wave computes a 16x64 output strip; K stepped by 4 via V_WMMA_F32_16X16X4_F32.
// A frag (16x4 f32, 2 VGPR): lane l (0..15) row m0+l; vgpr r holds K = k0 + 2*half + r.
// B frag (4x16  f32, 2 VGPR): vgpr r holds row K = k0 + r + 2*half, col n0+l.
// C/D (16x16 f32, 8 VGPR): vgpr r holds row m0 + r + 8*half, col n0+l.
__global__ void gemm_wmma_kernel(const float* __restrict__ X, const float* __restrict__ W,
                                 float* __restrict__ out, int M, int Nc) {
    const int lane = threadIdx.x & 31;
    const int wave = threadIdx.x >> 5;
    const int l    = lane & 15;
    const int half = lane >> 4;

    const int strips_per_row = Nc >> 6;                // 64-wide strips
    const int num_strips     = (M >> 4) * strips_per_row;
    const int strip          = blockIdx.x * (blockDim.x >> 5) + wave;
    if (strip >= num_strips) return;                   // wave-uniform: EXEC stays all-ones

    const int m0 = (strip / strips_per_row) << 4;
    const int n0 = (strip % strips_per_row) << 6;

    v8f acc0 = {}, acc1 = {}, acc2 = {}, acc3 = {};
    const float* xrow = X + (size_t)(m0 + l) * 128;

    for (int k0 = 0; k0 < 128; k0 += 4) {
        v2f a = *reinterpret_cast<const v2f*>(xrow + k0 + 2 * half);

        const float* w0 = W + (size_t)(k0 + 2 * half) * Nc + n0 + l;
        const float* w1 = w0 + Nc;
        v2f b0, b1, b2, b3;
        b0[0] = w0[0];  b0[1] = w1[0];
        b1[0] = w0[16]; b1[1] = w1[16];
        b2[0] = w0[32]; b2[1] = w1[32];
        b3[0] = w0[48]; b3[1] = w1[48];

        acc0 = __builtin_amdgcn_wmma_f32_16x16x4_f32(false, a, false, b0, (short)0, acc0, false, false);
        acc1 = __builtin_amdgcn_wmma_f32_16x16x4_f32(false, a, false, b1, (short)0, acc1, false, false);
        acc2 = __builtin_amdgcn_wmma_f32_16x16x4_f32(false, a, false, b2, (short)0, acc2, false, false);
        acc3 = __builtin_amdgcn_wmma_f32_16x16x4_f32(false, a, false, b3, (short)0, acc3, false, false);
    }

    float* orow = out + (size_t)(m0 + 8 * half) * Nc + n0 + l;
    #pragma unroll
    for (int r = 0; r < 8; ++r) {
        float* p = orow + (size_t)r * Nc;
        p[0]  = acc0[r];
        p[16] = acc1[r];
        p[32] = acc2[r];
        p[48] = acc3[r];
    }
}

// ---------------- edge scatter: agg[dst] += h[src] * dis[src]*dis[dst] ----------------
// One thread per (edge, float4-chunk). F = 4 << logC.
__global__ void scatter_kernel(const float* __restrict__ h, const long long* __restrict__ src,
                               const long long* __restrict__ dst, const float* __restrict__ dis,
                               float* __restrict__ agg, unsigned total, int logC) {
    unsigned tid = blockIdx.x * blockDim.x + threadIdx.x;
    if (tid >= total) return;
    const int F = 4 << logC;
    unsigned e = tid >> logC;
    unsigned c = tid & ((1u << logC) - 1);
    int s = (int)src[e];
    int d = (int)dst[e];
    float norm = dis[s] * dis[d];
    float4 v = reinterpret_cast<const float4*>(h + (size_t)s * F)[c];
    float* p = agg + (size_t)d * F + (size_t)c * 4;
    atomicAdd(p + 0, v.x * norm);
    atomicAdd(p + 1, v.y * norm);
    atomicAdd(p + 2, v.z * norm);
    atomicAdd(p + 3, v.w * norm);
}

// ---------------- finalize: out = agg + h*dis^2 + bias, optional relu ----------------
__global__ void finalize_kernel(const float* __restrict__ agg, const float* __restrict__ h,
                                const float* __restrict__ dis, const float* __restrict__ bias,
                                float* __restrict__ out, unsigned total, int logF, int do_relu) {
    unsigned tid = blockIdx.x * blockDim.x + threadIdx.x;
    if (tid >= total) return;
    unsigned i = tid >> logF;
    unsigned f = tid & ((1u << logF) - 1);
    float di = dis[i];
    float v  = agg[tid] + h[tid] * (di * di) + bias[f];
    if (do_relu) v = fmaxf(v, 0.0f);
    out[tid] = v;
}

// ---------------- host side ----------------

static inline int cdiv_i(int a, int b) { return (a + b - 1) / b; }

extern "C" void kernel_launch(void* const* d_in, const int* in_sizes, int n_in,
                              void* d_out, int out_size, void* d_ws, size_t ws_size,
                              hipStream_t stream) {
    (void)in_sizes; (void)n_in; (void)out_size; (void)ws_size;

    const float*     x    = (const float*)d_in[0];
    const long long* ei   = (const long long*)d_in[1];   // int64 edge_index [2,E]
    const float*     W1   = (const float*)d_in[2];
    const float*     b1   = (const float*)d_in[3];
    const float*     W2   = (const float*)d_in[4];
    const float*     b2   = (const float*)d_in[5];
    const float*     Wmu  = (const float*)d_in[6];
    const float*     bmu  = (const float*)d_in[7];
    const float*     Wls  = (const float*)d_in[8];
    const float*     bls  = (const float*)d_in[9];
    float*           out  = (float*)d_out;

    const int N = NNODES, E = NEDGES;
    const long long* srcp = ei;
    const long long* dstp = ei + E;

    // workspace layout (floats, 256B-aligned chunks)
    float* ws = (float*)d_ws;
    size_t o = 0;
    auto wsalloc = [&](size_t n) { float* p = ws + o; o += (n + 63) & ~(size_t)63; return p; };
    float* dis = wsalloc((size_t)N);          // degree -> deg_inv_sqrt (in place)
    float* A   = wsalloc((size_t)N * 128);
    float* B   = wsalloc((size_t)N * 128);
    float* Cb  = wsalloc((size_t)N * 64);
    float* Db  = wsalloc((size_t)N * 64);

    const dim3 thr(256);

    // ---- degrees: deg = 1 (self loop) + scatter ones; dis = rsqrt(deg)
    fill_kernel<<<cdiv_i(N, 256), thr, 0, stream>>>(dis, 1.0f, N);
    deg_kernel<<<cdiv_i(E, 256), thr, 0, stream>>>(dstp, dis, E);
    rsqrt_kernel<<<cdiv_i(N, 256), thr, 0, stream>>>(dis, N);

    auto gemm = [&](const float* Xp, const float* Wp, float* Op, int Nc) {
        int strips = (N / 16) * (Nc / 64);
        gemm_wmma_kernel<<<cdiv_i(strips, 4), dim3(128), 0, stream>>>(Xp, Wp, Op, N, Nc);
    };
    auto scatter = [&](const float* hp, float* aggp, int logF) {
        unsigned total = (unsigned)E << (logF - 2);     // chunks per edge = F/4
        scatter_kernel<<<cdiv_i((int)total, 256), thr, 0, stream>>>(hp, srcp, dstp, dis, aggp,
                                                                    total, logF - 2);
    };
    auto finalize = [&](const float* aggp, const float* hp, const float* bp, float* op,
                        int logF, int relu) {
        unsigned total = (unsigned)N << logF;
        finalize_kernel<<<cdiv_i((int)total, 256), thr, 0, stream>>>(aggp, hp, dis, bp, op,
                                                                     total, logF, relu);
    };

    // ---- layer 1: h1 = relu(agg(X@W1) + (X@W1)*dis^2 + b1)   (result -> A)
    gemm(x, W1, A, 128);
    fill_kernel<<<cdiv_i(N * 128, 256), thr, 0, stream>>>(B, 0.0f, N * 128);
    scatter(A, B, 7);
    finalize(B, A, b1, A, 7, 1);

    // ---- layer 2 (result -> B)
    gemm(A, W2, B, 128);
    fill_kernel<<<cdiv_i(N * 128, 256), thr, 0, stream>>>(A, 0.0f, N * 128);
    scatter(B, A, 7);
    finalize(A, B, b2, B, 7, 1);

    // ---- mu head -> d_out[0 : N*64]
    gemm(B, Wmu, Cb, 64);
    fill_kernel<<<cdiv_i(N * 64, 256), thr, 0, stream>>>(Db, 0.0f, N * 64);
    scatter(Cb, Db, 6);
    finalize(Db, Cb, bmu, out, 6, 0);

    // ---- logstd head -> d_out[N*64 : 2*N*64]
    gemm(B, Wls, Cb, 64);
    fill_kernel<<<cdiv_i(N * 64, 256), thr, 0, stream>>>(Db, 0.0f, N * 64);
    scatter(Cb, Db, 6);
    finalize(Db, Cb, bls, out + (size_t)N * 64, 6, 0);
}